// PhaseTransitionSmoother_23742579212586
// MI455X (gfx1250) — compile-verified
//
#include <hip/hip_runtime.h>

// EWMA linear recurrence s_t = 0.7*p_t + 0.3*s_{t-1} over (64, 16384, 32) f32.
// Chunked scan-as-matmul on CDNA5 WMMA (v_wmma_f32_16x16x4_f32): a constant
// 16x16 lower-triangular decay matrix times a 16-timestep x 16-channel tile,
// 4 chained K=4 WMMAs per tile. 48-step lookback (0.3^48 ~ 1e-25 << fp32 eps)
// makes every 512-step segment independent -> 4096 waves stream ~268 MB at
// HBM rate (~11.5 us roofline @ 23.3 TB/s).

typedef float v2f __attribute__((ext_vector_type(2)));
typedef float v8f __attribute__((ext_vector_type(8)));

#define EW_ALPHA 0.7f
#define EW_BETA  0.3f

__host__ __device__ constexpr float cbpow(int e) {
    float r = 1.0f;
    for (int i = 0; i < e; ++i) r *= EW_BETA;
    return r;
}

__global__ __launch_bounds__(256, 1)
void ewma_scan_wmma(const float* __restrict__ pred, float* __restrict__ out)
{
    constexpr int S    = 16384;
    constexpr int C    = 32;
    constexpr int TSEG = 512;           // stored timesteps per wave
    constexpr int NSEG = S / TSEG;      // 32 segments per batch
    constexpr int LOOK = 3;             // lookback tiles (3*16 = 48 steps)

    const int lane = threadIdx.x & 31;
    const int wid  = blockIdx.x * (blockDim.x >> 5) + (threadIdx.x >> 5);
    const int half = wid & 1;                 // which 16-channel half
    const int seg  = (wid >> 1) & (NSEG - 1); // segment along S
    const int b    = wid / (2 * NSEG);        // batch index
    const int n    = lane & 15;               // channel column / matrix row (M=N index)
    const int hi   = lane >> 4;               // lane half (0/1)
    const int k0   = hi * 2;                  // K base row for this lane half
    const int c0   = half * 16;

    // ---- per-lane constants, all folded from compile-time powers -----------
    // BETA^n via binary decomposition of n (4 predicated multiplies)
    float bn = 1.0f;
    if (n & 1) bn *= cbpow(1);
    if (n & 2) bn *= cbpow(2);
    if (n & 4) bn *= cbpow(4);
    if (n & 8) bn *= cbpow(8);

    // A (decay) matrix, M[m][t'] = (t'<=m) ? ALPHA*BETA^(m-t') : 0, m = n.
    // A 16x4 layout: VGPR0 holds K=k0, VGPR1 holds K=k0+1.
    // ALPHA*BETA^(n-t') = (ALPHA/BETA^t') * BETA^n with the first factor constexpr.
    v2f a0, a1, a2, a3;
    {
        const float inv2 = (k0 == 2) ? 1.0f / cbpow(2) : 1.0f;  // fold k0 into constant
#define EW_W(tp) ((n >= (tp) + k0) ? (EW_ALPHA / cbpow(tp)) * inv2 * bn : 0.0f)
        a0.x = EW_W(0);  a0.y = EW_W(1);
        a1.x = EW_W(4);  a1.y = EW_W(5);
        a2.x = EW_W(8);  a2.y = EW_W(9);
        a3.x = EW_W(12); a3.y = EW_W(13);
#undef EW_W
    }

    // carry factors: D VGPR v holds row m = v + hi*8, weight BETA^(m+1)
    const float hb = hi ? cbpow(8) : 1.0f;
    float f[8];
#pragma unroll
    for (int v = 0; v < 8; ++v) f[v] = cbpow(v + 1) * hb;

    // ---- segment setup -----------------------------------------------------
    const int startT     = (seg == 0) ? 0 : seg * TSEG - LOOK * 16;
    const int tiles      = (seg == 0) ? TSEG / 16 : TSEG / 16 + LOOK;
    const int firstStore = (seg == 0) ? 0 : LOOK;

    const size_t segBase = ((size_t)b * S + (size_t)startT) * C + c0;
    const float* ptr  = pred + segBase + n + (size_t)k0 * C;      // B-slab loads
    float*       optr = out  + segBase + n + (size_t)hi * 8 * C;  // D stores

    // Segment 0: carry = p_0  =>  s_0 = 0.7*p_0 + 0.3*p_0 = p_0 (reference init).
    float carry = (seg == 0) ? pred[(size_t)b * S * C + c0 + n] : 0.0f;

    // preload first tile's B slabs (row r of the tile lives at ptr[r*C])
    v2f nb0, nb1, nb2, nb3;
    nb0.x = ptr[0 * C];  nb0.y = ptr[1 * C];
    nb1.x = ptr[4 * C];  nb1.y = ptr[5 * C];
    nb2.x = ptr[8 * C];  nb2.y = ptr[9 * C];
    nb3.x = ptr[12 * C]; nb3.y = ptr[13 * C];

    for (int t = 0; t < tiles; ++t) {
        v2f b0 = nb0, b1 = nb1, b2 = nb2, b3 = nb3;
        ptr += 16 * C;
        if (t + 1 < tiles) {   // prefetch next tile past the WMMAs
            nb0.x = ptr[0 * C];  nb0.y = ptr[1 * C];
            nb1.x = ptr[4 * C];  nb1.y = ptr[5 * C];
            nb2.x = ptr[8 * C];  nb2.y = ptr[9 * C];
            nb3.x = ptr[12 * C]; nb3.y = ptr[13 * C];
        }

        // D = M(16x16) @ P(16x16) via 4 chained 16x16x4 f32 WMMAs
        v8f acc = {};
        acc = __builtin_amdgcn_wmma_f32_16x16x4_f32(false, a0, false, b0, (short)0, acc, false, false);
        acc = __builtin_amdgcn_wmma_f32_16x16x4_f32(false, a1, false, b1, (short)0, acc, false, false);
        acc = __builtin_amdgcn_wmma_f32_16x16x4_f32(false, a2, false, b2, (short)0, acc, false, false);
        acc = __builtin_amdgcn_wmma_f32_16x16x4_f32(false, a3, false, b3, (short)0, acc, false, false);

        // add decayed carry: s_row += BETA^(row+1) * carry[channel]
#pragma unroll
        for (int v = 0; v < 8; ++v) acc[v] = fmaf(f[v], carry, acc[v]);

        if (t >= firstStore) {   // wave-uniform branch, EXEC stays full
#pragma unroll
            for (int v = 0; v < 8; ++v)
                __builtin_nontemporal_store(acc[v], &optr[v * C]);  // write-once output
        }
        optr += 16 * C;

        // next carry = s_15[channel n], held in D VGPR7 lanes 16..31
        carry = __shfl(acc[7], 16 + n, 32);
    }
}

extern "C" void kernel_launch(void* const* d_in, const int* in_sizes, int n_in,
                              void* d_out, int out_size, void* d_ws, size_t ws_size,
                              hipStream_t stream) {
    (void)in_sizes; (void)n_in; (void)out_size; (void)d_ws; (void)ws_size;
    const float* pred = (const float*)d_in[0];
    float* out = (float*)d_out;

    constexpr int B = 64, S = 16384, TSEG = 512;
    constexpr int totalWaves = B * (S / TSEG) * 2;   // 4096 wave32s
    dim3 block(256);                                  // 8 waves per block
    dim3 grid(totalWaves / 8);                        // 512 blocks
    ewma_scan_wmma<<<grid, block, 0, stream>>>(pred, out);
}